// VectorQuantizer_76570676953198
// MI455X (gfx1250) — compile-verified
//
#include <hip/hip_runtime.h>
#include <hip/hip_bf16.h>
#include <math.h>

// ---------------------------------------------------------------------------
// VQ-VAE vector quantizer for MI455X (gfx1250, wave32).
//   z:      (32, 64, 64, 64) f32   -> N = 131072 rows of D = 64
//   weight: (512, 64) f32          -> K = 512 codes
// Outputs (concatenated f32): loss[1], z_q[8388608], perplexity[1],
//                             encodings[67108864], codebook[131072]
//
// Roofline: ~335 MB of streaming traffic (268 MB one-hot) vs 8.6 GFLOP GEMM
// -> memory bound; f32 WMMA 16x16x4 handles the score GEMM natively (argmin
// is precision-sensitive, so we stay in the reference's fp32), NT hints keep
// the streaming outputs out of the caches so the codebook stays resident.
// ---------------------------------------------------------------------------

#define VQ_N      131072      // 32*64*64 rows
#define VQ_D      64
#define VQ_K      512
#define VQ_HW     4096        // H*W
#define VQ_CHW    262144      // C*H*W
#define VQ_ZELEMS 8388608     // 32*64*64*64
#define VQ_BETA   0.25f
#define VQ_EPS    1e-10f

typedef __attribute__((ext_vector_type(2))) float v2f;
typedef __attribute__((ext_vector_type(8))) float v8f;

// d_out layout (floats)
#define OFF_LOSS  0
#define OFF_ZQ    1
#define OFF_PERP  8388609
#define OFF_ENC   8388610
#define OFF_CB    75497474

// d_ws layout (bytes)
#define WS_CBI    0           // int[131072]
#define WS_ENORM  524288      // float[512]
#define WS_HIST   526336      // float[512]
#define WS_LOSS   528384      // float[1]

// ---------------------------------------------------------------------------
// Kernel 1: per-code squared norms, zero hist + loss accumulator.
// One block of 512 threads.
// ---------------------------------------------------------------------------
__global__ void vq_init_kernel(const float* __restrict__ weight,
                               float* __restrict__ enorm,
                               float* __restrict__ hist,
                               float* __restrict__ loss_acc) {
    int k = threadIdx.x;                 // 0..511
    const float* wr = weight + k * VQ_D;
    float s = 0.0f;
#pragma unroll
    for (int d = 0; d < VQ_D; ++d) {
        float w = wr[d];
        s = fmaf(w, w, s);
    }
    enorm[k] = s;
    hist[k]  = 0.0f;
    if (k == 0) loss_acc[0] = 0.0f;
}

// ---------------------------------------------------------------------------
// Kernel 2: argmin_k ( ||e_k||^2 - 2 z.e_k ) per row via f32 WMMA.
// One wave (32 lanes) = one 16-row tile. 128 threads/block = 4 waves.
// A fragment (16x4 f32): lanes 0-15 rows M=0..15 {K=0,K=1}; lanes 16-31 {K=2,K=3}.
// C/D (16x16 f32): VGPR v -> row v (lanes 0-15) / row v+8 (lanes 16-31), col = lane&15.
// ---------------------------------------------------------------------------
__global__ void vq_argmin_wmma_kernel(const float* __restrict__ z,
                                      const float* __restrict__ weight,
                                      const float* __restrict__ enorm,
                                      float* __restrict__ cb_f,     // d_out codebook (float)
                                      int*   __restrict__ cb_i,     // ws codebook (int)
                                      float* __restrict__ hist) {
    const int lane = threadIdx.x & 31;
    const int wave = threadIdx.x >> 5;
    const int row0 = (blockIdx.x * 4 + wave) * 16;   // 16-row tile base (n index)

    // ---- preload all 16 A fragments (this wave's 16x64 z-rows) -------------
    // zf[n][c] = z[(n>>12)*CHW + c*HW + (n & 4095)]
    const int   lrow  = lane & 15;
    const int   choff = (lane >> 4) << 1;            // 0 (lanes 0-15) or 2 (16-31)
    const float* zbase = z + (size_t)(row0 >> 12) * VQ_CHW + (row0 & 4095) + lrow;

    v2f afrag[16];
#pragma unroll
    for (int kk = 0; kk < 16; ++kk) {
        int c0 = 4 * kk + choff;
        afrag[kk][0] = zbase[(size_t)c0       * VQ_HW];   // coalesced across lanes 0-15
        afrag[kk][1] = zbase[(size_t)(c0 + 1) * VQ_HW];
    }

    // ---- sweep 32 code tiles of 16 codes -----------------------------------
    float bv[8];
    int   bi[8];
#pragma unroll
    for (int v = 0; v < 8; ++v) { bv[v] = INFINITY; bi[v] = 0; }

    for (int kt = 0; kt < VQ_K / 16; ++kt) {
        const int cbase = kt * 16;
        const float* wrow = weight + (size_t)(cbase + lrow) * VQ_D + choff;

        v8f acc = {0.f, 0.f, 0.f, 0.f, 0.f, 0.f, 0.f, 0.f};
#pragma unroll
        for (int kk = 0; kk < 16; ++kk) {
            v2f bfrag = *reinterpret_cast<const v2f*>(wrow + 4 * kk);
            acc = __builtin_amdgcn_wmma_f32_16x16x4_f32(
                /*neg_a=*/false, afrag[kk],
                /*neg_b=*/false, bfrag,
                /*c_mod=*/(short)0, acc,
                /*reuse_a=*/false, /*reuse_b=*/false);
        }

        const int   code = cbase + lrow;
        const float en   = enorm[code];
#pragma unroll
        for (int v = 0; v < 8; ++v) {
            float dist = fmaf(-2.0f, acc[v], en);
            if (dist < bv[v]) { bv[v] = dist; bi[v] = code; }
        }
    }

    // ---- min-reduce (val,idx) across the 16 lanes of each half -------------
#pragma unroll
    for (int v = 0; v < 8; ++v) {
        float bvv = bv[v];
        int   bii = bi[v];
#pragma unroll
        for (int off = 1; off < 16; off <<= 1) {
            float ov = __shfl_xor(bvv, off, 32);
            int   oi = __shfl_xor(bii, off, 32);
            if (ov < bvv || (ov == bvv && oi < bii)) { bvv = ov; bii = oi; }
        }
        if (lrow == 0) {
            int m = v + ((lane >> 4) << 3);          // lane0 -> rows 0..7, lane16 -> 8..15
            int n = row0 + m;
            cb_f[n] = (float)bii;
            cb_i[n] = bii;
            atomicAdd(&hist[bii], 1.0f);
        }
    }
}

// ---------------------------------------------------------------------------
// Kernel 3: z_q gather (back to B,C,H,W) + loss partial sums.
// t indexes z_q/z directly; consecutive t -> consecutive w (coalesced).
// NT store for z_q (write-once), NT load for z (last use); wave32 shuffle
// reduction then a single LDS combine per block.
// ---------------------------------------------------------------------------
__global__ void vq_zq_loss_kernel(const float* __restrict__ z,
                                  const float* __restrict__ weight,
                                  const int*   __restrict__ cb_i,
                                  float* __restrict__ zq,
                                  float* __restrict__ loss_acc) {
    const int t   = blockIdx.x * blockDim.x + threadIdx.x;   // 0..8388607
    const int pos = t & 4095;
    const int c   = (t >> 12) & 63;
    const int b   = t >> 18;
    const int n   = (b << 12) | pos;

    const int   idx = cb_i[n];
    const float qv  = weight[idx * VQ_D + c];
    __builtin_nontemporal_store(qv, &zq[t]);
    float d  = qv - __builtin_nontemporal_load(&z[t]);
    float sq = d * d;

    // wave32 tree reduction
#pragma unroll
    for (int off = 16; off >= 1; off >>= 1)
        sq += __shfl_xor(sq, off, 32);

    __shared__ float red[8];                                  // 8 waves / block
    const int wave = threadIdx.x >> 5;
    const int lane = threadIdx.x & 31;
    if (lane == 0) red[wave] = sq;
    __syncthreads();
    if (wave == 0) {
        float s = (lane < 8) ? red[lane] : 0.0f;
#pragma unroll
        for (int off = 4; off >= 1; off >>= 1)
            s += __shfl_xor(s, off, 32);
        if (lane == 0) atomicAdd(loss_acc, s);
    }
}

// ---------------------------------------------------------------------------
// Kernel 4: one-hot encodings. Each thread emits 8 floats as 4x b64 NT
// stores (268 MB total; section is only 8-byte aligned so b64 is the widest
// legal store). 64 threads cover one 512-wide row.
// ---------------------------------------------------------------------------
__global__ void vq_encodings_kernel(const int* __restrict__ cb_i,
                                    v2f* __restrict__ enc) {
    const int g  = blockIdx.x * blockDim.x + threadIdx.x;    // 0..8388607 groups
    const int k0 = (g & 63) << 3;                            // first k of 8-float group
    const int n  = g >> 6;
    const int idx = cb_i[n];
    v2f* p = enc + ((size_t)g << 2);
#pragma unroll
    for (int j = 0; j < 4; ++j) {
        v2f v;
        v[0] = (k0 + 2 * j     == idx) ? 1.0f : 0.0f;
        v[1] = (k0 + 2 * j + 1 == idx) ? 1.0f : 0.0f;
        __builtin_nontemporal_store(v, &p[j]);
    }
}

// ---------------------------------------------------------------------------
// Kernel 5: finalize loss + perplexity. One block of 512 threads.
// ---------------------------------------------------------------------------
__global__ void vq_finalize_kernel(const float* __restrict__ hist,
                                   const float* __restrict__ loss_acc,
                                   float* __restrict__ out_loss,
                                   float* __restrict__ out_perp) {
    __shared__ float red[512];
    const int k = threadIdx.x;
    float p = hist[k] * (1.0f / (float)VQ_N);
    red[k] = p * logf(p + VQ_EPS);
    __syncthreads();
#pragma unroll
    for (int s = 256; s > 0; s >>= 1) {
        if (k < s) red[k] += red[k + s];
        __syncthreads();
    }
    if (k == 0) {
        out_perp[0] = expf(-red[0]);
        // loss = mean((zq - z)^2) + BETA * mean((zq - z)^2) (stop_gradient is value-neutral)
        out_loss[0] = loss_acc[0] * ((1.0f + VQ_BETA) / (float)VQ_ZELEMS);
    }
}

// ---------------------------------------------------------------------------
extern "C" void kernel_launch(void* const* d_in, const int* in_sizes, int n_in,
                              void* d_out, int out_size, void* d_ws, size_t ws_size,
                              hipStream_t stream) {
    const float* z      = (const float*)d_in[0];
    const float* weight = (const float*)d_in[1];
    float* out = (float*)d_out;

    char* ws = (char*)d_ws;
    int*   cb_i     = (int*)  (ws + WS_CBI);
    float* enorm    = (float*)(ws + WS_ENORM);
    float* hist     = (float*)(ws + WS_HIST);
    float* loss_acc = (float*)(ws + WS_LOSS);

    float*  out_loss = out + OFF_LOSS;
    float*  zq       = out + OFF_ZQ;
    float*  out_perp = out + OFF_PERP;
    v2f*    enc      = (v2f*)(out + OFF_ENC);
    float*  cb_f     = out + OFF_CB;

    vq_init_kernel<<<1, 512, 0, stream>>>(weight, enorm, hist, loss_acc);

    // 131072 rows / 16 rows-per-wave / 4 waves-per-block = 2048 blocks
    vq_argmin_wmma_kernel<<<2048, 128, 0, stream>>>(z, weight, enorm,
                                                    cb_f, cb_i, hist);

    vq_zq_loss_kernel<<<VQ_ZELEMS / 256, 256, 0, stream>>>(z, weight, cb_i,
                                                           zq, loss_acc);

    // 131072 rows * 64 groups of 8 floats = 8,388,608 threads
    vq_encodings_kernel<<<8388608 / 256, 256, 0, stream>>>(cb_i, enc);

    vq_finalize_kernel<<<1, 512, 0, stream>>>(hist, loss_acc,
                                              out_loss, out_perp);
}